// QuantumGate_65274912964829
// MI455X (gfx1250) — compile-verified
//
#include <hip/hip_runtime.h>
#include <hip/hip_bf16.h>
#include <math.h>

// ---------------------------------------------------------------------------
// 12-qubit circuit simulator, one batch element per 256-thread block (8 wave32).
// State (4096 complex) lives in LDS. Per layer, the 24 single-qubit gates are
// fused into three 16x16 complex unitaries (Kron of four 2x2 Rot*RY) and
// applied with V_WMMA_F32_16X16X4_F32 (complex matmul = 4 real WMMAs; the
// minus sign of Gr*Sr - Gi*Si is realized by negating the imag A-fragment,
// since f32 WMMA only supports C-negate per the ISA NEG table).
// The CNOT ring is one GF(2)-linear index permutation per layer.
// ---------------------------------------------------------------------------

typedef float v2f __attribute__((ext_vector_type(2)));
typedef float v8f __attribute__((ext_vector_type(8)));

#define NQ   12
#define QDIM 4096
#define NLAY 2

static __device__ __forceinline__ v8f wmma4(v2f a, v2f b, v8f c) {
  // D(16x16,f32) = A(16x4,f32) x B(4x16,f32) + C
  return __builtin_amdgcn_wmma_f32_16x16x4_f32(false, a, false, b,
                                               (short)0, c, false, false);
}

__global__ void __launch_bounds__(256)
qsim12_wmma_kernel(const float* __restrict__ x, const float* __restrict__ w,
                   float* __restrict__ out, int batch) {
  __shared__ float sRe[QDIM];
  __shared__ float sIm[QDIM];
  __shared__ float gRe[256];   // fused 16x16 group gate, row-major
  __shared__ float gIm[256];
  __shared__ float mRe[16];    // four 2x2 wire matrices [s][i][j]
  __shared__ float mIm[16];
  __shared__ float zsum[16];

  const int t    = threadIdx.x;
  const int b    = blockIdx.x;
  const int lane = t & 31;
  const int wv   = t >> 5;      // wave id 0..7
  const int l16  = lane & 15;
  const int hi   = lane >> 4;   // half-wave: selects K pair (kk+2*hi, +1)
  if (b >= batch) return;

  // |00...0>
  for (int k = 0; k < 16; ++k) {
    int i = t + 256 * k;
    sRe[i] = (i == 0) ? 1.0f : 0.0f;
    sIm[i] = 0.0f;
  }
  __syncthreads();

  for (int l = 0; l < NLAY; ++l) {
    for (int g = 0; g < 3; ++g) {
      // ---- per-wire 2x2 matrices: M = Rot(phi,theta,omega) * RY(x) ----
      if (t < 4) {
        int q = 4 * g + t;
        float sx, cx; __sincosf(0.5f * x[b * NQ + q], &sx, &cx);
        const float* wq = w + (l * NQ + q) * 3;
        float phi = wq[0], th = wq[1], om = wq[2];
        float st, ct; __sincosf(0.5f * th, &st, &ct);
        float sp, cp; __sincosf(0.5f * (phi + om), &sp, &cp);
        float sm, cm; __sincosf(0.5f * (phi - om), &sm, &cm);
        float Rr[2][2], Ri[2][2];
        Rr[0][0] =  cp * ct; Ri[0][0] = -sp * ct;   // e^{-i(phi+om)/2} c
        Rr[0][1] = -cm * st; Ri[0][1] = -sm * st;   // -e^{ i(phi-om)/2} s
        Rr[1][0] =  cm * st; Ri[1][0] = -sm * st;   //  e^{-i(phi-om)/2} s
        Rr[1][1] =  cp * ct; Ri[1][1] =  sp * ct;   //  e^{ i(phi+om)/2} c
        // RY = [[cx,-sx],[sx,cx]]
        #pragma unroll
        for (int i2 = 0; i2 < 2; ++i2) {
          mRe[t * 4 + i2 * 2 + 0] =  Rr[i2][0] * cx + Rr[i2][1] * sx;
          mIm[t * 4 + i2 * 2 + 0] =  Ri[i2][0] * cx + Ri[i2][1] * sx;
          mRe[t * 4 + i2 * 2 + 1] = -Rr[i2][0] * sx + Rr[i2][1] * cx;
          mIm[t * 4 + i2 * 2 + 1] = -Ri[i2][0] * sx + Ri[i2][1] * cx;
        }
      }
      __syncthreads();

      // ---- G = M0 (x) M1 (x) M2 (x) M3 ; wire 4g maps to bit 3 of i/j ----
      {
        int gi = t >> 4, gj = t & 15;
        float pr = 1.0f, pim = 0.0f;
        #pragma unroll
        for (int s = 0; s < 4; ++s) {
          int ib = (gi >> (3 - s)) & 1;
          int jb = (gj >> (3 - s)) & 1;
          float mr = mRe[s * 4 + ib * 2 + jb];
          float mi = mIm[s * 4 + ib * 2 + jb];
          float nr = pr * mr - pim * mi;
          float ni = pr * mi + pim * mr;
          pr = nr; pim = ni;
        }
        gRe[t] = pr; gIm[t] = pim;
      }
      __syncthreads();

      // ---- apply G via WMMA; waves own disjoint state slices (no barrier) --
      if (g == 0) {
        // wires 0-3 = top index bits p: new = G x S, S[p][c], idx=(p<<8)|c
        #pragma unroll
        for (int tt = 0; tt < 2; ++tt) {
          int c = ((wv + 8 * tt) << 4) | l16;
          v8f cr = {}; v8f ci = {};
          #pragma unroll
          for (int kk = 0; kk < 16; kk += 4) {
            int c0 = kk + 2 * hi;
            v2f ar, ai, ain, br, bi;
            ar.x = gRe[l16 * 16 + c0]; ar.y = gRe[l16 * 16 + c0 + 1];
            ai.x = gIm[l16 * 16 + c0]; ai.y = gIm[l16 * 16 + c0 + 1];
            ain.x = -ai.x; ain.y = -ai.y;
            br.x = sRe[(c0 << 8) | c]; br.y = sRe[((c0 + 1) << 8) | c];
            bi.x = sIm[(c0 << 8) | c]; bi.y = sIm[((c0 + 1) << 8) | c];
            cr = wmma4(ar, br, cr);  cr = wmma4(ain, bi, cr);
            ci = wmma4(ar, bi, ci);  ci = wmma4(ai, br, ci);
          }
          #pragma unroll
          for (int v = 0; v < 8; ++v) {
            int m = v + 8 * hi;
            sRe[(m << 8) | c] = cr[v];
            sIm[(m << 8) | c] = ci[v];
          }
        }
      } else if (g == 1) {
        // wires 4-7 = middle bits: per top-4-bit slice a, new = G x S[a]
        #pragma unroll
        for (int tt = 0; tt < 2; ++tt) {
          int abase = (wv + 8 * tt) << 8;
          v8f cr = {}; v8f ci = {};
          #pragma unroll
          for (int kk = 0; kk < 16; kk += 4) {
            int c0 = kk + 2 * hi;
            v2f ar, ai, ain, br, bi;
            ar.x = gRe[l16 * 16 + c0]; ar.y = gRe[l16 * 16 + c0 + 1];
            ai.x = gIm[l16 * 16 + c0]; ai.y = gIm[l16 * 16 + c0 + 1];
            ain.x = -ai.x; ain.y = -ai.y;
            br.x = sRe[abase | (c0 << 4) | l16];
            br.y = sRe[abase | ((c0 + 1) << 4) | l16];
            bi.x = sIm[abase | (c0 << 4) | l16];
            bi.y = sIm[abase | ((c0 + 1) << 4) | l16];
            cr = wmma4(ar, br, cr);  cr = wmma4(ain, bi, cr);
            ci = wmma4(ar, bi, ci);  ci = wmma4(ai, br, ci);
          }
          #pragma unroll
          for (int v = 0; v < 8; ++v) {
            int m = v + 8 * hi;
            sRe[abase | (m << 4) | l16] = cr[v];
            sIm[abase | (m << 4) | l16] = ci[v];
          }
        }
      } else {
        // wires 8-11 = low bits p: new = S x G^T, S[r][p], idx=(r<<4)|p.
        // B-fragment of G^T per-lane equals A-fragment of G.
        #pragma unroll
        for (int tt = 0; tt < 2; ++tt) {
          int rbase = (wv + 8 * tt) << 4;
          int r = rbase + l16;
          v8f cr = {}; v8f ci = {};
          #pragma unroll
          for (int kk = 0; kk < 16; kk += 4) {
            int c0 = kk + 2 * hi;
            v2f ar, ai, ain, bgr, bgi;
            ar.x = sRe[(r << 4) | c0]; ar.y = sRe[(r << 4) | (c0 + 1)];
            ai.x = sIm[(r << 4) | c0]; ai.y = sIm[(r << 4) | (c0 + 1)];
            ain.x = -ai.x; ain.y = -ai.y;
            bgr.x = gRe[l16 * 16 + c0]; bgr.y = gRe[l16 * 16 + c0 + 1];
            bgi.x = gIm[l16 * 16 + c0]; bgi.y = gIm[l16 * 16 + c0 + 1];
            cr = wmma4(ar, bgr, cr);  cr = wmma4(ain, bgi, cr);
            ci = wmma4(ar, bgi, ci);  ci = wmma4(ai, bgr, ci);
          }
          #pragma unroll
          for (int v = 0; v < 8; ++v) {
            int m = v + 8 * hi;
            sRe[((rbase + m) << 4) | l16] = cr[v];
            sIm[((rbase + m) << 4) | l16] = ci[v];
          }
        }
      }
      __syncthreads();
    }

    // ---- CNOT ring = single linear bit permutation: amp_new[f(i)] = amp[i].
    // wire q <-> bit (11-q). f: bit(11-t) = XOR of bits 11..11-t (t>=1),
    // bit 11 = XOR of bits 10..0. Suffix-XOR via shift cascade.
    {
      float tr[16], ti[16];
      #pragma unroll
      for (int k = 0; k < 16; ++k) {
        int i = t + 256 * k;
        tr[k] = sRe[i]; ti[k] = sIm[i];
      }
      __syncthreads();
      #pragma unroll
      for (int k = 0; k < 16; ++k) {
        unsigned i = (unsigned)(t + 256 * k);
        unsigned y = i;
        y ^= y >> 1; y ^= y >> 2; y ^= y >> 4; y ^= y >> 8;
        unsigned j = (y & 0x7FFu) | (((y ^ (i >> 11)) & 1u) << 11);
        sRe[j] = tr[k]; sIm[j] = ti[k];
      }
      __syncthreads();
    }
  }

  // ---- <Z_q> = sum probs * (+1 if bit(11-q)==0 else -1) ----
  float acc[NQ];
  #pragma unroll
  for (int q = 0; q < NQ; ++q) acc[q] = 0.0f;
  for (int k = 0; k < 16; ++k) {
    int i = t + 256 * k;
    float re = sRe[i], im = sIm[i];
    float p = re * re + im * im;
    #pragma unroll
    for (int q = 0; q < NQ; ++q)
      acc[q] += ((i >> (11 - q)) & 1) ? -p : p;
  }
  // wave32 reduction, then cross-wave via LDS atomics (ds_add_f32)
  #pragma unroll
  for (int q = 0; q < NQ; ++q) {
    float v = acc[q];
    for (int off = 16; off > 0; off >>= 1) v += __shfl_down(v, off, 32);
    acc[q] = v;
  }
  if (t < NQ) zsum[t] = 0.0f;
  __syncthreads();
  if (lane == 0) {
    for (int q = 0; q < NQ; ++q) atomicAdd(&zsum[q], acc[q]);
  }
  __syncthreads();
  if (t < NQ) out[b * NQ + t] = zsum[t];
}

extern "C" void kernel_launch(void* const* d_in, const int* in_sizes, int n_in,
                              void* d_out, int out_size, void* d_ws, size_t ws_size,
                              hipStream_t stream) {
  (void)n_in; (void)out_size; (void)d_ws; (void)ws_size;
  const float* x = (const float*)d_in[0];      // (B, 12) f32
  const float* w = (const float*)d_in[1];      // (2, 12, 3) f32
  float* out = (float*)d_out;                  // (B, 12) f32
  int batch = in_sizes[0] / NQ;
  qsim12_wmma_kernel<<<batch, 256, 0, stream>>>(x, w, out, batch);
}